// GaussianMoments4_60748017435175
// MI455X (gfx1250) — compile-verified
//
#include <hip/hip_runtime.h>

// ---------------------------------------------------------------------------
// GaussianMoments4 on MI455X (gfx1250, wave32)
//   m2/m3/m4 cluster moments as per-wave SGEMMs via V_WMMA_F32_16X16X4_F32.
//   m3/m4 staging uses async global->LDS copies (ASYNCcnt) + double buffering.
// ---------------------------------------------------------------------------

typedef __attribute__((ext_vector_type(2))) float v2f;
typedef __attribute__((ext_vector_type(8))) float v8f;

#define Bn 1024
#define Cn 10
#define Dn 16
#define EPSF 1e-7f
#define CHUNKS (Bn / 32)

__device__ __forceinline__ v8f wmma4(v2f a, v2f b, v8f c) {
  // D = A(16x4, f32) * B(4x16, f32) + C(16x16, f32)
  return __builtin_amdgcn_wmma_f32_16x16x4_f32(
      /*neg_a=*/false, a, /*neg_b=*/false, b,
      /*c_mod=*/(short)0, c, /*reuse_a=*/false, /*reuse_b=*/false);
}

// --- async global->LDS copy helpers (gfx1250 ASYNCcnt path) -----------------
__device__ __forceinline__ void asyncLdsB64(uint32_t ldsOff, const void* g) {
  asm volatile("global_load_async_to_lds_b64 %0, %1, off"
               :: "v"(ldsOff), "v"((uint64_t)(uintptr_t)g) : "memory");
}
__device__ __forceinline__ void asyncLdsB128(uint32_t ldsOff, const void* g) {
  asm volatile("global_load_async_to_lds_b128 %0, %1, off"
               :: "v"(ldsOff), "v"((uint64_t)(uintptr_t)g) : "memory");
}
__device__ __forceinline__ void waitAsync1() {
  asm volatile("s_wait_asynccnt 0x1" ::: "memory");  // keep 1 copy in flight
}
__device__ __forceinline__ void waitAsync0() {
  asm volatile("s_wait_asynccnt 0x0" ::: "memory");
}

// stage one 32-sample x 16-float chunk of dd into an LDS buffer.
// 256 threads x b64: thread t covers sample (t>>3), elements 2*(t&7)..+1.
__device__ __forceinline__ void stageChunk(const float* dd, int chunk,
                                           uint32_t ldsBase, int t) {
  int elemOff = (t >> 3) * Dn + (t & 7) * 2;
  asyncLdsB64(ldsBase + (uint32_t)(elemOff * 4), dd + chunk * 32 * Dn + elemOff);
}

// signed compressive roots: sign(sign(x)+0.1) * ((|x|+c1)^p - c2)
__device__ __forceinline__ float sroot2(float x) {
  float s = (x >= 0.f) ? 1.f : -1.f;
  return s * (sqrtf(fabsf(x) + 0.25f) - 0.5f);
}
__device__ __forceinline__ float sroot3(float x) {
  float s = (x >= 0.f) ? 1.f : -1.f;
  return s * (cbrtf(fabsf(x) + 0.19245008973f) - 0.57735026919f);
}
__device__ __forceinline__ float sroot4(float x) {
  float s = (x >= 0.f) ? 1.f : -1.f;
  return s * (sqrtf(sqrtf(fabsf(x) + 0.15749013123f)) - 0.62996052494f);
}

__device__ __forceinline__ float waveSum(float v) {
#pragma unroll
  for (int off = 16; off > 0; off >>= 1) v += __shfl_xor(v, off, 32);
  return v;
}

// --- zero the accumulators (harness poisons d_out / d_ws) -------------------
__global__ void k_init(float* out, int out_n, float* cw, float* sumd) {
  int t = threadIdx.x;
  if (t < out_n) out[t] = 0.f;
  if (t < 16) cw[t] = 0.f;
  if (t < Cn * Dn) sumd[t] = 0.f;
}

// --- per-sample: argmax assignment, d row, cluster count, sum(d) ------------
__global__ void k_prep(const float* __restrict__ emb, const float* __restrict__ cen,
                       const float* __restrict__ lg, int* __restrict__ assign,
                       float* __restrict__ dd, float* cw, float* sumd) {
  int b = blockIdx.x * 256 + threadIdx.x;  // grid covers exactly Bn
  float best = lg[b * Cn];
  int a = 0;
#pragma unroll
  for (int c = 1; c < Cn; ++c) {
    float v = lg[b * Cn + c];
    if (v > best) { best = v; a = c; }   // ties -> first index, matches argmax
  }
  assign[b] = a;
  atomicAdd(&cw[a], 1.0f);
#pragma unroll
  for (int i = 0; i < Dn; ++i) {
    float v = emb[b * Dn + i] - cen[a * Dn + i];
    dd[b * Dn + i] = v;
    atomicAdd(&sumd[a * Dn + i], v);
  }
}

// --- p1 term (first moment) -------------------------------------------------
__global__ void k_p1(const float* gm1, const float* mw1, const float* cw,
                     const float* sumd, float* out) {
  int t = threadIdx.x;
  if (t >= Cn * Dn) return;
  int c = t >> 4, i = t & 15;
  float cwc = cw[c];
  float m1 = sumd[t] / (cwc + EPSF);
  float w = (cwc * (1.0f / Bn)) * mw1[i];
  float dl = m1 - gm1[i];
  atomicAdd(out, 1.0f * w * dl * dl);
}

// --- m2: per-cluster 16x16 = D^T D, one wave per cluster (sync staging) -----
__global__ void k_m2(const int* __restrict__ assign, const float* __restrict__ dd,
                     const float* cw, const float* __restrict__ gm2,
                     const float* __restrict__ mw2, float* out) {
  __shared__ float sdm[32 * Dn];
  const int c = blockIdx.x;
  const int l = threadIdx.x;
  const int lo = l & 15;
  const int hiK = (l >> 4) * 2;  // A layout: lanes0-15 K=0/1, lanes16-31 K=2/3
  v8f acc = {};
  for (int base = 0; base < Bn; base += 32) {
    int s = base + l;
    float m = (assign[s] == c) ? 1.0f : 0.0f;  // mask folded once (0/1 idempotent)
#pragma unroll
    for (int i = 0; i < Dn; ++i) sdm[l * Dn + i] = dd[s * Dn + i] * m;
    __syncthreads();
#pragma unroll
    for (int k = 0; k < 32; k += 4) {
      int s0 = k + hiK, s1 = s0 + 1;
      v2f A; A.x = sdm[s0 * Dn + lo]; A.y = sdm[s1 * Dn + lo];
      acc = wmma4(A, A, acc);  // symmetric: B fragment identical
    }
    __syncthreads();
  }
  float cwc = cw[c];
  float inv = 1.0f / (cwc + EPSF);
  float cwn = cwc * (1.0f / Bn);
  int miB = (l >> 4) << 3;  // C/D layout: M = r + 8*(lane>=16)
  float part = 0.f;
#pragma unroll
  for (int r = 0; r < 8; ++r) {
    int flat = (r + miB) * Dn + lo;
    float v = sroot2(acc[r] * inv);
    float tv = sroot2(gm2[flat]);
    float w = cwn * mw2[flat];
    float dl = v - tv;
    part += w * dl * dl;
  }
  part = waveSum(part * 0.5f);
  if (l == 0) atomicAdd(out, part);
}

// --- m3: per-cluster 16x256 = D^T E; async double-buffered staging ----------
__global__ void __launch_bounds__(256)
k_m3(const int* __restrict__ assign, const float* __restrict__ dd,
     const float* cw, const float* __restrict__ gm3,
     const float* __restrict__ mw3, float* out) {
  __shared__ float sbuf[2][32 * Dn];
  __shared__ int sasg[Bn];
  const int c = blockIdx.x >> 1;
  const int ntHalf = blockIdx.x & 1;
  const int t = threadIdx.x;
  const int wave = t >> 5, lane = t & 31;
  const int nt = ntHalf * 8 + wave;  // n-tile (j index of e = d_j * d_k)
  const int lo = lane & 15, hiK = (lane >> 4) * 2;

  // stage full assignment table once (4 ints/thread), then chunk 0
  asyncLdsB128((uint32_t)(uintptr_t)&sasg[t * 4], assign + t * 4);
  stageChunk(dd, 0, (uint32_t)(uintptr_t)&sbuf[0][0], t);

  v8f acc = {};
  for (int i = 0; i < CHUNKS; ++i) {
    const int p = i & 1;
    if (i + 1 < CHUNKS) {
      stageChunk(dd, i + 1, (uint32_t)(uintptr_t)&sbuf[p ^ 1][0], t);
      waitAsync1();  // in-order: chunk i (and sasg) complete, i+1 in flight
    } else {
      waitAsync0();
    }
    __syncthreads();
    const float* sdm = sbuf[p];
    const int cb = i * 32;
#pragma unroll
    for (int k = 0; k < 32; k += 4) {
      int s0 = k + hiK, s1 = s0 + 1;
      float m0 = (sasg[cb + s0] == c) ? 1.0f : 0.0f;  // mask^1 on A is exact
      float m1 = (sasg[cb + s1] == c) ? 1.0f : 0.0f;
      float d0 = sdm[s0 * Dn + lo], d1 = sdm[s1 * Dn + lo];
      v2f A;  A.x = m0 * d0;               A.y = m1 * d1;               // A[i,K]=d_i
      v2f Bv; Bv.x = sdm[s0 * Dn + nt] * d0; Bv.y = sdm[s1 * Dn + nt] * d1; // B=d_j*d_k
      acc = wmma4(A, Bv, acc);
    }
    __syncthreads();  // all waves done with sbuf[p] before it is re-staged
  }
  float cwn = cw[c] * (1.0f / Bn);
  int miB = (lane >> 4) << 3;
  float part = 0.f;
#pragma unroll
  for (int r = 0; r < 8; ++r) {
    int flat = (r + miB) * 256 + nt * Dn + lo;  // (i,j,k) flat
    float v = sroot3(acc[r]);                   // no count division (per ref)
    float tv = sroot3(gm3[flat]);
    float w = cwn * mw3[flat];
    float dl = v - tv;
    part += w * dl * dl;
  }
  part = waveSum(part * 0.25f);
  if (lane == 0) atomicAdd(out, part);
}

// --- m4: per-cluster 256x256 = E^T E; async double-buffered staging ---------
__global__ void __launch_bounds__(256)
k_m4(const int* __restrict__ assign, const float* __restrict__ dd,
     const float* cw, const float* __restrict__ gm4,
     const float* __restrict__ mw4, float* out) {
  __shared__ float sbuf[2][32 * Dn];
  __shared__ int sasg[Bn];
  const int bid = blockIdx.x;
  const int ntHalf = bid & 1;
  const int mt = (bid >> 1) & 15;
  const int c = bid >> 5;
  const int t = threadIdx.x;
  const int wave = t >> 5, lane = t & 31;
  const int nt = ntHalf * 8 + wave;
  const int lo = lane & 15, hiK = (lane >> 4) * 2;

  asyncLdsB128((uint32_t)(uintptr_t)&sasg[t * 4], assign + t * 4);
  stageChunk(dd, 0, (uint32_t)(uintptr_t)&sbuf[0][0], t);

  v8f acc = {};
  for (int i = 0; i < CHUNKS; ++i) {
    const int p = i & 1;
    if (i + 1 < CHUNKS) {
      stageChunk(dd, i + 1, (uint32_t)(uintptr_t)&sbuf[p ^ 1][0], t);
      waitAsync1();
    } else {
      waitAsync0();
    }
    __syncthreads();
    const float* sdm = sbuf[p];
    const int cb = i * 32;
#pragma unroll
    for (int k = 0; k < 32; k += 4) {
      int s0 = k + hiK, s1 = s0 + 1;
      float m0 = (sasg[cb + s0] == c) ? 1.0f : 0.0f;
      float m1 = (sasg[cb + s1] == c) ? 1.0f : 0.0f;
      float d0 = sdm[s0 * Dn + lo], d1 = sdm[s1 * Dn + lo];
      v2f A;  A.x = m0 * sdm[s0 * Dn + mt] * d0;  A.y = m1 * sdm[s1 * Dn + mt] * d1; // e[m]
      v2f Bv; Bv.x = sdm[s0 * Dn + nt] * d0;      Bv.y = sdm[s1 * Dn + nt] * d1;     // e[n]
      acc = wmma4(A, Bv, acc);
    }
    __syncthreads();
  }
  float cwn = cw[c] * (1.0f / Bn);
  int miB = (lane >> 4) << 3;
  float part = 0.f;
#pragma unroll
  for (int r = 0; r < 8; ++r) {
    int mi = r + miB;
    int flat = (mt * Dn + mi) * 256 + nt * Dn + lo;  // (i,j,k,l) flat
    float v = sroot4(acc[r]);
    float tv = sroot4(gm4[flat]);
    float w = cwn * mw4[flat];
    float dl = v - tv;
    part += w * dl * dl;
  }
  part = waveSum(part * 0.125f);
  if (lane == 0) atomicAdd(out, part);
}

extern "C" void kernel_launch(void* const* d_in, const int* in_sizes, int n_in,
                              void* d_out, int out_size, void* d_ws, size_t ws_size,
                              hipStream_t stream) {
  const float* emb = (const float*)d_in[0];
  const float* cen = (const float*)d_in[1];
  const float* lg  = (const float*)d_in[2];
  const float* gm1 = (const float*)d_in[3];
  const float* gm2 = (const float*)d_in[4];
  const float* gm3 = (const float*)d_in[5];
  const float* gm4 = (const float*)d_in[6];
  const float* mw1 = (const float*)d_in[7];
  const float* mw2 = (const float*)d_in[8];
  const float* mw3 = (const float*)d_in[9];
  const float* mw4 = (const float*)d_in[10];
  float* out = (float*)d_out;

  float* W = (float*)d_ws;
  int*   assign = (int*)W;          // 1024 ints
  float* dd   = W + 1024;           // 1024*16 floats
  float* cw   = W + 1024 + Bn * Dn; // 16 floats (10 used)
  float* sumd = cw + 16;            // 160 floats

  k_init<<<1, 256, 0, stream>>>(out, out_size, cw, sumd);
  k_prep<<<Bn / 256, 256, 0, stream>>>(emb, cen, lg, assign, dd, cw, sumd);
  k_p1  <<<1, 256, 0, stream>>>(gm1, mw1, cw, sumd, out);
  k_m2  <<<Cn, 32, 0, stream>>>(assign, dd, cw, gm2, mw2, out);
  k_m3  <<<Cn * 2, 256, 0, stream>>>(assign, dd, cw, gm3, mw3, out);
  k_m4  <<<Cn * 16 * 2, 256, 0, stream>>>(assign, dd, cw, gm4, mw4, out);
}